// RAG_secondary_retrieval_10024453669301
// MI455X (gfx1250) — compile-verified
//
#include <hip/hip_runtime.h>
#include <hip/hip_bf16.h>
#include <math.h>

#define NVOX  32768   // 32*32*32 voxels (B=1)
#define M_DB  4096
#define TOPK  10

typedef float v2f __attribute__((ext_vector_type(2)));
typedef float v8f __attribute__((ext_vector_type(8)));
typedef int   v4i __attribute__((vector_size(16)));   // GCC-style vector, matches builtin sig
typedef v4i __attribute__((address_space(1)))* gv4i_p;
typedef v4i __attribute__((address_space(3)))* lv4i_p;

#if defined(__has_builtin)
#if __has_builtin(__builtin_amdgcn_global_load_async_to_lds_b128)
#define HAVE_ASYNC_LDS 1
#endif
#endif

// ---------------- conv1: 2 -> 16, k3, pad1, + BN + ReLU ----------------
__global__ void k_conv1(const float* __restrict__ bg, const float* __restrict__ ed,
                        const float* __restrict__ w1, const float* __restrict__ b1,
                        const float* __restrict__ g1, const float* __restrict__ be1,
                        const float* __restrict__ m1, const float* __restrict__ v1,
                        float* __restrict__ x1) {
    int v = blockIdx.x * blockDim.x + threadIdx.x;
    if (v >= NVOX) return;
    int d = v >> 10, h = (v >> 5) & 31, w = v & 31;
    float acc[16];
#pragma unroll
    for (int c = 0; c < 16; ++c) acc[c] = 0.f;
    for (int dz = -1; dz <= 1; ++dz) {
        int dd = d + dz; if ((unsigned)dd >= 32u) continue;
        for (int dy = -1; dy <= 1; ++dy) {
            int hh = h + dy; if ((unsigned)hh >= 32u) continue;
            for (int dx = -1; dx <= 1; ++dx) {
                int ww = w + dx; if ((unsigned)ww >= 32u) continue;
                int nv = (dd << 10) | (hh << 5) | ww;
                int t  = (dz + 1) * 9 + (dy + 1) * 3 + (dx + 1);
                float i0 = bg[nv], i1 = ed[nv];
#pragma unroll
                for (int co = 0; co < 16; ++co)
                    acc[co] += w1[co * 54 + t] * i0 + w1[co * 54 + 27 + t] * i1;
            }
        }
    }
#pragma unroll
    for (int co = 0; co < 16; ++co) {
        float s = g1[co] * rsqrtf(v1[co] + 1e-5f);
        float y = (acc[co] + b1[co] - m1[co]) * s + be1[co];
        x1[v * 16 + co] = fmaxf(y, 0.f);
    }
}

// ------- conv2: 16 -> 32, k3, pad1, + BN + ReLU; co split over blockIdx.y -------
__global__ void k_conv2(const float* __restrict__ x1, const float* __restrict__ w2,
                        const float* __restrict__ b2, const float* __restrict__ g2,
                        const float* __restrict__ be2, const float* __restrict__ m2,
                        const float* __restrict__ v2, float* __restrict__ x2) {
    __shared__ float sw[16 * 16 * 27];            // 27,648 B: this block's co-slice
    const int cob = blockIdx.y * 16;              // 0 or 16
    for (int i = threadIdx.x; i < 16 * 16 * 27; i += blockDim.x)
        sw[i] = w2[cob * 432 + i];                // w2 is co-major contiguous
    __syncthreads();
    int v = blockIdx.x * blockDim.x + threadIdx.x;
    if (v >= NVOX) return;
    int d = v >> 10, h = (v >> 5) & 31, w = v & 31;
    float acc[16];
#pragma unroll
    for (int c = 0; c < 16; ++c) acc[c] = 0.f;
    for (int dz = -1; dz <= 1; ++dz) {
        int dd = d + dz; if ((unsigned)dd >= 32u) continue;
        for (int dy = -1; dy <= 1; ++dy) {
            int hh = h + dy; if ((unsigned)hh >= 32u) continue;
            for (int dx = -1; dx <= 1; ++dx) {
                int ww = w + dx; if ((unsigned)ww >= 32u) continue;
                int nv = (dd << 10) | (hh << 5) | ww;
                int t  = (dz + 1) * 9 + (dy + 1) * 3 + (dx + 1);
                for (int ci = 0; ci < 16; ++ci) {
                    float xv = x1[nv * 16 + ci];
                    const float* wp = sw + ci * 27 + t;
#pragma unroll
                    for (int co = 0; co < 16; ++co)
                        acc[co] += wp[co * 432] * xv;
                }
            }
        }
    }
#pragma unroll
    for (int co = 0; co < 16; ++co) {
        int c = cob + co;
        float s = g2[c] * rsqrtf(v2[c] + 1e-5f);
        float y = (acc[co] + b2[c] - m2[c]) * s + be2[c];
        x2[v * 32 + c] = fmaxf(y, 0.f);
    }
}

// ---------------- conv3 (1x1, 32 -> 8) + L2 normalize -> q[N][8] ----------------
__global__ void k_conv3(const float* __restrict__ x2, const float* __restrict__ w3,
                        const float* __restrict__ b3, float* __restrict__ q) {
    int v = blockIdx.x * blockDim.x + threadIdx.x;
    if (v >= NVOX) return;
    float lat[8];
#pragma unroll
    for (int c = 0; c < 8; ++c) lat[c] = b3[c];
    for (int ci = 0; ci < 32; ++ci) {
        float xv = x2[v * 32 + ci];
#pragma unroll
        for (int c = 0; c < 8; ++c) lat[c] += w3[c * 32 + ci] * xv;
    }
    float ss = 0.f;
#pragma unroll
    for (int c = 0; c < 8; ++c) ss += lat[c] * lat[c];
    float inv = 1.f / fmaxf(sqrtf(ss), 1e-12f);
#pragma unroll
    for (int c = 0; c < 8; ++c) q[v * 8 + c] = lat[c] * inv;
}

__device__ __forceinline__ void topk_insert(float (&bd)[TOPK], int (&bi)[TOPK],
                                            float sc, int id) {
    if (sc > bd[TOPK - 1]) {
        bd[TOPK - 1] = sc; bi[TOPK - 1] = id;
#pragma unroll
        for (int t = TOPK - 1; t > 0; --t) {
            if (bd[t] > bd[t - 1]) {
                float tf = bd[t]; bd[t] = bd[t - 1]; bd[t - 1] = tf;
                int   ti = bi[t]; bi[t] = bi[t - 1]; bi[t - 1] = ti;
            }
        }
    }
}

// ---------------- kNN: WMMA dot-product GEMM + split top-10 ----------------
// db (128 KB) staged once per block into LDS (async global->LDS when available).
// Each wave owns 16 queries; per round: 32 db columns, 4x V_WMMA_F32_16X16X4_F32
// (K=8 exact fp32). Scores -> LDS; lane L scans cols 0..15, lane L+16 scans
// cols 16..31 of its query row (all 32 lanes busy), private top-10 each,
// merged exactly at the end. d2 = 2 - 2*dot (unit-norm q, db).
__global__ void k_knn(const float* __restrict__ q, const float* __restrict__ db,
                      const float* __restrict__ lbl, float* __restrict__ out) {
    __shared__ __align__(16) float sdb[M_DB * 8];   // 131,072 B
    __shared__ float st[8 * 16 * 33];               //  16,896 B (per-wave 16x33 tiles)
    const int lane = threadIdx.x & 31;
    const int wv   = threadIdx.x >> 5;
    const int m0   = (blockIdx.x * 8 + wv) * 16;
    const int mn   = lane & 15;           // query row within tile / col within sub-tile
    const int hi   = lane >> 4;           // 0: low half-wave, 1: high half-wave
    const int kb   = hi * 2;              // K sub-block per ISA 32-bit A/B layout
    float* s = st + wv * (16 * 33);

    // ---- stage db into LDS, coalesced 16B per lane ----
#if HAVE_ASYNC_LDS
    for (int j = threadIdx.x; j < (M_DB * 8) / 4; j += 256)
        __builtin_amdgcn_global_load_async_to_lds_b128(
            (gv4i_p)(db + j * 4), (lv4i_p)(sdb + j * 4), 0, 0);
#if __has_builtin(__builtin_amdgcn_s_wait_asynccnt)
    __builtin_amdgcn_s_wait_asynccnt(0);
#else
    asm volatile("s_wait_asynccnt 0" ::: "memory");
#endif
#else
    for (int j = threadIdx.x; j < (M_DB * 8) / 4; j += 256)
        ((float4*)sdb)[j] = ((const float4*)db)[j];
#endif
    __syncthreads();

    // A fragments (fp32 16x4 layout: lanes 0-15 K=kb,kb+1; lanes 16-31 K=kb+2.. )
    v2f a0 = *(const v2f*)(q + (m0 + mn) * 8 + kb);       // K-frag 0..3
    v2f a1 = *(const v2f*)(q + (m0 + mn) * 8 + 4 + kb);   // K-frag 4..7

    float bdot[TOPK]; int bidx[TOPK];
#pragma unroll
    for (int k = 0; k < TOPK; ++k) { bdot[k] = -1e30f; bidx[k] = 0; }

    for (int n0 = 0; n0 < M_DB; n0 += 32) {
        const float* p0 = sdb + (n0 + mn) * 8 + kb;        // B tile A (cols n0..n0+15)
        const float* p1 = sdb + (n0 + 16 + mn) * 8 + kb;   // B tile B (cols +16..+31)
        v2f b0a = *(const v2f*)(p0);
        v2f b1a = *(const v2f*)(p0 + 4);
        v2f b0b = *(const v2f*)(p1);
        v2f b1b = *(const v2f*)(p1 + 4);
        v8f cA = {0.f,0.f,0.f,0.f,0.f,0.f,0.f,0.f};
        v8f cB = {0.f,0.f,0.f,0.f,0.f,0.f,0.f,0.f};
        cA = __builtin_amdgcn_wmma_f32_16x16x4_f32(false, a0, false, b0a, (short)0, cA, false, false);
        cB = __builtin_amdgcn_wmma_f32_16x16x4_f32(false, a0, false, b0b, (short)0, cB, false, false);
        cA = __builtin_amdgcn_wmma_f32_16x16x4_f32(false, a1, false, b1a, (short)0, cA, false, false);
        cB = __builtin_amdgcn_wmma_f32_16x16x4_f32(false, a1, false, b1b, (short)0, cB, false, false);
        // C layout: VGPR r -> (row = r + 8*hi, col = mn); stride-33 rows kill conflicts
        int rbase = hi * 8;
#pragma unroll
        for (int r = 0; r < 8; ++r) {
            s[(rbase + r) * 33 + mn]      = cA[r];
            s[(rbase + r) * 33 + 16 + mn] = cB[r];
        }
        __syncthreads();
        {   // all 32 lanes scan: query row = mn, column half = hi*16
            const float* row = s + mn * 33 + hi * 16;
            const int nb = n0 + hi * 16;
#pragma unroll
            for (int j = 0; j < 16; ++j)
                topk_insert(bdot, bidx, row[j], nb + j);
        }
        __syncthreads();
    }

    // ---- merge high-half top-10 into low-half top-10 via LDS ----
    if (hi) {
#pragma unroll
        for (int k = 0; k < TOPK; ++k) {
            s[mn * 20 + k]         = bdot[k];
            s[mn * 20 + TOPK + k]  = __int_as_float(bidx[k]);
        }
    }
    __syncthreads();
    if (!hi) {
#pragma unroll
        for (int k = 0; k < TOPK; ++k)
            topk_insert(bdot, bidx, s[mn * 20 + k],
                        __float_as_int(s[mn * 20 + TOPK + k]));
        float wsum = 1e-8f, ps = 0.f;
#pragma unroll
        for (int k = 0; k < TOPK; ++k) {
            float d2 = 2.f - 2.f * bdot[k];
            float wk = __expf(-10.f * d2);
            ps   += wk * lbl[bidx[k]];
            wsum += wk;
        }
        out[m0 + mn] = ps / wsum;
    }
}

extern "C" void kernel_launch(void* const* d_in, const int* in_sizes, int n_in,
                              void* d_out, int out_size, void* d_ws, size_t ws_size,
                              hipStream_t stream) {
    const float* bg  = (const float*)d_in[0];
    const float* ed  = (const float*)d_in[1];
    // d_in[2] = context_mask (unused by reference math)
    const float* w1  = (const float*)d_in[3];
    const float* b1  = (const float*)d_in[4];
    const float* g1  = (const float*)d_in[5];
    const float* be1 = (const float*)d_in[6];
    const float* m1  = (const float*)d_in[7];
    const float* v1  = (const float*)d_in[8];
    const float* w2  = (const float*)d_in[9];
    const float* b2  = (const float*)d_in[10];
    const float* g2  = (const float*)d_in[11];
    const float* be2 = (const float*)d_in[12];
    const float* m2  = (const float*)d_in[13];
    const float* v2  = (const float*)d_in[14];
    const float* w3  = (const float*)d_in[15];
    const float* b3  = (const float*)d_in[16];
    const float* db  = (const float*)d_in[17];
    const float* lbl = (const float*)d_in[18];
    float* out = (float*)d_out;

    float* x1 = (float*)d_ws;            // [NVOX][16]  2 MB
    float* x2 = x1 + (size_t)NVOX * 16;  // [NVOX][32]  4 MB
    float* qv = x2 + (size_t)NVOX * 32;  // [NVOX][8]   1 MB

    k_conv1<<<NVOX / 256, 256, 0, stream>>>(bg, ed, w1, b1, g1, be1, m1, v1, x1);
    k_conv2<<<dim3(NVOX / 256, 2), 256, 0, stream>>>(x1, w2, b2, g2, be2, m2, v2, x2);
    k_conv3<<<NVOX / 256, 256, 0, stream>>>(x2, w3, b3, qv);
    k_knn<<<NVOX / (16 * 8), 256, 0, stream>>>(qv, db, lbl, out);
}